// GCN_68049461837860
// MI455X (gfx1250) — compile-verified
//
#include <hip/hip_runtime.h>
#include <cmath>

typedef float v2f __attribute__((ext_vector_type(2)));
typedef float v8f __attribute__((ext_vector_type(8)));

#define F 128   // F_IN == HID == 128
#define NGR 64  // N_GRAPHS

// ---------------- utility -------------------------------------------------
__global__ void fill_kernel(float* __restrict__ p, float v, long n) {
  long i = (long)blockIdx.x * blockDim.x + threadIdx.x;
  if (i < n) p[i] = v;
}

// deg[dst] += ew  (deg pre-filled with 1.0 for the self-loop)
__global__ void deg_kernel(const int* __restrict__ dst, const float* __restrict__ ew,
                           float* __restrict__ deg, int E) {
  int e = blockIdx.x * blockDim.x + threadIdx.x;
  if (e < E) atomicAdd(&deg[dst[e]], ew[e]);
}

// in-place deg -> dinv
__global__ void dinv_kernel(float* __restrict__ deg, int n) {
  int i = blockIdx.x * blockDim.x + threadIdx.x;
  if (i < n) { float d = deg[i]; deg[i] = (d > 0.f) ? rsqrtf(d) : 0.f; }
}

// norm[e] = dinv[src]*ew*dinv[dst]
__global__ void norm_kernel(const int* __restrict__ src, const int* __restrict__ dst,
                            const float* __restrict__ ew, const float* __restrict__ dinv,
                            float* __restrict__ norm, int E) {
  int e = blockIdx.x * blockDim.x + threadIdx.x;
  if (e < E) norm[e] = dinv[src[e]] * ew[e] * dinv[dst[e]];
}

// ---------------- fp32 WMMA GEMM: H[n,128] = X[n,128] @ W[128,128] --------
// One wave computes a 16-row strip across all 8 column tiles.
// W staged once per workgroup in LDS (64 KB).
__global__ __launch_bounds__(256)
void gemm_wmma_f32(const float* __restrict__ X, const float* __restrict__ W,
                   float* __restrict__ H, int n) {
  __shared__ float sW[F * F];
  int tid = threadIdx.x;
  // cooperative stage of W (128x128 fp32 = 64KB) via float4
  for (int i = tid; i < F * F / 4; i += 256)
    ((float4*)sW)[i] = ((const float4*)W)[i];
  __syncthreads();

  int wave = tid >> 5;
  int lane = tid & 31;
  long strip = (long)blockIdx.x * 8 + wave;      // 16-row strip index
  long row0  = strip * 16;
  if (row0 >= n) return;

  int m     = lane & 15;   // row within strip (both half-waves)
  int khalf = lane >> 4;   // 0: K={kc,kc+1}; 1: K={kc+2,kc+3}
  long rowm = row0 + m;
  if (rowm >= n) rowm = n - 1;                   // safe clamp (n % 16 == 0 here)
  const float* xrow = X + rowm * F;

  v8f acc[8];
#pragma unroll
  for (int t = 0; t < 8; ++t) acc[t] = (v8f)0.f;

  for (int kc = 0; kc < F; kc += 4) {
    int k = kc + khalf * 2;
    v2f a;
    a.x = xrow[k];
    a.y = xrow[k + 1];
#pragma unroll
    for (int t = 0; t < 8; ++t) {
      int nn = t * 16 + (lane & 15);
      v2f b;
      b.x = sW[k * F + nn];          // B fragment: K rows striped like A
      b.y = sW[(k + 1) * F + nn];
      acc[t] = __builtin_amdgcn_wmma_f32_16x16x4_f32(
          /*neg_a=*/false, a, /*neg_b=*/false, b,
          /*c_mod=*/(short)0, acc[t], /*reuse_a=*/false, /*reuse_b=*/false);
    }
  }

  // C/D layout: VGPR r -> (M=r, lanes 0-15) and (M=r+8, lanes 16-31), N = lane&15
#pragma unroll
  for (int t = 0; t < 8; ++t) {
#pragma unroll
    for (int r = 0; r < 8; ++r) {
      long mm = row0 + r + khalf * 8;
      if (mm < n) H[mm * F + t * 16 + (lane & 15)] = acc[t][r];
    }
  }
}

// ---------------- edge scatter: OUT[dst] += norm[e] * H[src] --------------
// 8 edges per 256-thread block; 32 lanes/edge; 4 feats/lane (float4 gather).
__global__ __launch_bounds__(256)
void scatter_kernel(const int* __restrict__ src, const int* __restrict__ dst,
                    const float* __restrict__ norm, const float* __restrict__ H,
                    float* __restrict__ OUT, int E) {
  int e = blockIdx.x * 8 + (threadIdx.x >> 5);
  if (e >= E) return;
  int lane = threadIdx.x & 31;
  int s = src[e], d = dst[e];
  float w = norm[e];
  float4 hv = ((const float4*)(H + (long)s * F))[lane];
  float* o = OUT + (long)d * F + lane * 4;
  atomicAdd(o + 0, w * hv.x);
  atomicAdd(o + 1, w * hv.y);
  atomicAdd(o + 2, w * hv.z);
  atomicAdd(o + 3, w * hv.w);
}

// B[i] = ELU( B[i] + dinv[node]^2 * A[i] + bias[f] )   (self-loop fused)
__global__ void fuse_elu_kernel(float* __restrict__ B, const float* __restrict__ A,
                                const float* __restrict__ dinv,
                                const float* __restrict__ bias, long total) {
  long i = (long)blockIdx.x * blockDim.x + threadIdx.x;
  if (i >= total) return;
  long node = i >> 7;
  int f = (int)(i & 127);
  float di = dinv[node];
  float v = B[i] + di * di * A[i] + bias[f];
  B[i] = (v > 0.f) ? v : (expf(v) - 1.f);
}

// v = B + dinv^2*A + bias; sums[batch[node]][f] += v; counts[batch[node]] += 1
__global__ void fuse_pool_kernel(const float* __restrict__ B, const float* __restrict__ A,
                                 const float* __restrict__ dinv,
                                 const float* __restrict__ bias,
                                 const int* __restrict__ batch,
                                 float* __restrict__ sums, float* __restrict__ counts,
                                 long total) {
  long i = (long)blockIdx.x * blockDim.x + threadIdx.x;
  if (i >= total) return;
  long node = i >> 7;
  int f = (int)(i & 127);
  float di = dinv[node];
  float v = B[i] + di * di * A[i] + bias[f];
  int g = batch[node];
  atomicAdd(&sums[(long)g * F + f], v);
  if (f == 0) atomicAdd(&counts[g], 1.f);
}

// out[g] = dot(sums[g]/max(counts[g],1), Wfc) + bfc  — one wave per graph
__global__ __launch_bounds__(64)
void final_kernel(const float* __restrict__ sums, const float* __restrict__ counts,
                  const float* __restrict__ Wfc, const float* __restrict__ bfc,
                  float* __restrict__ out) {
  int g = blockIdx.x * 2 + (threadIdx.x >> 5);
  if (g >= NGR) return;
  int lane = threadIdx.x & 31;
  float c = fmaxf(counts[g], 1.f);
  float acc = 0.f;
  for (int f = lane; f < F; f += 32) acc += (sums[(long)g * F + f] / c) * Wfc[f];
  for (int off = 16; off > 0; off >>= 1) acc += __shfl_xor(acc, off, 32);
  if (lane == 0) out[g] = acc + bfc[0];
}

// ---------------- host orchestration --------------------------------------
extern "C" void kernel_launch(void* const* d_in, const int* in_sizes, int n_in,
                              void* d_out, int out_size, void* d_ws, size_t ws_size,
                              hipStream_t stream) {
  const float* x    = (const float*)d_in[0];
  const int*   ei   = (const int*)d_in[1];
  const float* ew   = (const float*)d_in[2];
  const int*   batch= (const int*)d_in[3];
  const float* W1   = (const float*)d_in[4];
  const float* b1   = (const float*)d_in[5];
  const float* W2   = (const float*)d_in[6];
  const float* b2   = (const float*)d_in[7];
  const float* Wfc  = (const float*)d_in[8];
  const float* bfc  = (const float*)d_in[9];
  float* out = (float*)d_out;

  const int n = in_sizes[3];        // N_NODES (batch is [n])
  const int E = in_sizes[2];        // N_EDGES (edge_weight is [E])
  const int* src = ei;
  const int* dst = ei + E;

  // workspace carve (floats)
  float* ws    = (float*)d_ws;
  float* dinv  = ws;                         // n
  float* nrm   = dinv + n;                   // E
  float* bufA  = nrm + E;                    // n*F
  float* bufB  = bufA + (long)n * F;         // n*F
  float* sums  = bufB + (long)n * F;         // NGR*F
  float* cnts  = sums + NGR * F;             // NGR
  (void)ws_size; (void)n_in; (void)out_size;

  const long totNF = (long)n * F;
  const int  TB = 256;
  dim3 blk(TB);
  int  gridE   = (E + TB - 1) / TB;
  int  gridN   = (n + TB - 1) / TB;
  long gridNF  = (totNF + TB - 1) / TB;
  int  gridSc  = (E + 7) / 8;                 // 8 edges / block
  int  strips  = (n + 15) / 16;
  int  gridG   = (strips + 7) / 8;            // 8 waves / block

  // 1) degrees (init 1.0 for self-loop) -> dinv
  fill_kernel<<<gridN, blk, 0, stream>>>(dinv, 1.0f, n);
  deg_kernel<<<gridE, blk, 0, stream>>>(dst, ew, dinv, E);
  dinv_kernel<<<gridN, blk, 0, stream>>>(dinv, n);
  // 2) edge norms
  norm_kernel<<<gridE, blk, 0, stream>>>(src, dst, ew, dinv, nrm, E);

  // ---- layer 1 ----
  gemm_wmma_f32<<<gridG, blk, 0, stream>>>(x, W1, bufA, n);
  fill_kernel<<<(int)gridNF, blk, 0, stream>>>(bufB, 0.0f, totNF);
  scatter_kernel<<<gridSc, blk, 0, stream>>>(src, dst, nrm, bufA, bufB, E);
  fuse_elu_kernel<<<(int)gridNF, blk, 0, stream>>>(bufB, bufA, dinv, b1, totNF);

  // ---- layer 2 ----
  gemm_wmma_f32<<<gridG, blk, 0, stream>>>(bufB, W2, bufA, n);
  fill_kernel<<<(int)gridNF, blk, 0, stream>>>(bufB, 0.0f, totNF);
  scatter_kernel<<<gridSc, blk, 0, stream>>>(src, dst, nrm, bufA, bufB, E);

  // ---- pool + fc ----
  fill_kernel<<<(NGR * F + NGR + TB - 1) / TB, blk, 0, stream>>>(sums, 0.0f, NGR * F + NGR);
  fuse_pool_kernel<<<(int)gridNF, blk, 0, stream>>>(bufB, bufA, dinv, b2, batch,
                                                    sums, cnts, totNF);
  final_kernel<<<NGR / 2, dim3(64), 0, stream>>>(sums, cnts, Wfc, bfc, out);
}